// ExpressionTree_53506702573658
// MI455X (gfx1250) — compile-verified
//
#include <hip/hip_runtime.h>

// CDNA5 / gfx1250: wave32, f32 WMMA 16x16x4.
typedef __attribute__((ext_vector_type(2))) float v2f;
typedef __attribute__((ext_vector_type(8))) float v8f;

#define D_DIM 1024
#define WAVES_PER_BLOCK 8
#define ROWS_PER_WAVE 16
#define BLOCK_THREADS (32 * WAVES_PER_BLOCK)

__global__ __launch_bounds__(BLOCK_THREADS)
void ExpressionTree_wmma_kernel(const float* __restrict__ x,
                                const float* __restrict__ c1,
                                const float* __restrict__ c2,
                                const float* __restrict__ w1p,
                                const float* __restrict__ b1p,
                                const float* __restrict__ w2p,
                                const float* __restrict__ b2p,
                                const float* __restrict__ wrp,
                                const float* __restrict__ brp,
                                float* __restrict__ out)
{
    const int lane = threadIdx.x & 31;
    const int wave = threadIdx.x >> 5;
    const int m    = lane & 15;           // A-matrix row this lane feeds / batch row in B
    const int koff = (lane >> 4) * 2;     // which half of the K=4 slice this lane-half holds

    const long row_base = ((long)blockIdx.x * WAVES_PER_BLOCK + wave) * ROWS_PER_WAVE;
    const float* xrow = x + (row_base + m) * (long)D_DIM + koff;

    // Uniform scalar parameters (single-element device arrays).
    const float w1 = *w1p, b1 = *b1p;
    const float w2 = *w2p, b2 = *b2p;
    const float wr = *wrp, br = *brp;

    // A-matrix (16x4): row 0 = c1 chunk, row 1 = c2 chunk, rows 2..15 = 0.
    // Single combined coefficient pointer: lanes m>=2 harmlessly alias c1 and
    // are zeroed by one hoisted predicate -> 2 loads + 4 cndmasks per chunk
    // instead of 4 loads + 8 cndmasks.
    const float* cptr  = (m == 1) ? c2 : c1;
    const bool   a_live = (m < 2);

    v8f acc0 = {0.f, 0.f, 0.f, 0.f, 0.f, 0.f, 0.f, 0.f};
    v8f acc1 = {0.f, 0.f, 0.f, 0.f, 0.f, 0.f, 0.f, 0.f};

    // K loop: 8 columns of x per iteration, 2 independent WMMA accumulators
    // (breaks the WMMA->WMMA RAW chain on the accumulator).
    #pragma unroll 2
    for (int k = 0; k < D_DIM; k += 8) {
        // B (4x16 f32, K x N): vgpr j, lane L -> K = (L/16)*2 + j, N = L%16.
        v2f bA = *(const v2f*)(xrow + k);        // K slice [k   .. k+3]
        v2f bB = *(const v2f*)(xrow + k + 4);    // K slice [k+4 .. k+7]

        // A (16x4 f32): vgpr j, lane L -> M = L%16, K = (L/16)*2 + j.
        v2f cA = *(const v2f*)(cptr + k + koff);
        v2f cB = *(const v2f*)(cptr + k + 4 + koff);

        v2f aA, aB;
        aA.x = a_live ? cA.x : 0.f;
        aA.y = a_live ? cA.y : 0.f;
        aB.x = a_live ? cB.x : 0.f;
        aB.y = a_live ? cB.y : 0.f;

        // D = A x B + C : D[0][n] += sum_k c1[k]*x[n][k]; D[1][n] += c2 dot.
        acc0 = __builtin_amdgcn_wmma_f32_16x16x4_f32(
                   false, aA, false, bA, (short)0, acc0, false, false);
        acc1 = __builtin_amdgcn_wmma_f32_16x16x4_f32(
                   false, aB, false, bB, (short)0, acc1, false, false);

        // Stream-ahead prefetch of x (global_prefetch_b8); speculative, safe at tail.
        __builtin_prefetch(xrow + k + 256, 0, 1);
    }

    // D layout: vgpr r holds M=r on lanes 0-15 (N = lane). For lanes 0..15:
    //   l1 (row M=0) = acc[0] on own lane, l2 (row M=1) = acc[1] on own lane.
    float l1 = acc0[0] + acc1[0];
    float l2 = acc0[1] + acc1[1];

    if (lane < 16) {
        float u1 = w1 * sinf(l1) + b1;
        float u2 = w2 * fmaxf(l2, 0.f) + b2;
        float v  = u1 * u2;
        out[row_base + lane] = wr * tanhf(v) + br;
    }
}

extern "C" void kernel_launch(void* const* d_in, const int* in_sizes, int n_in,
                              void* d_out, int out_size, void* d_ws, size_t ws_size,
                              hipStream_t stream) {
    const float* x  = (const float*)d_in[0];
    const float* c1 = (const float*)d_in[1];
    const float* c2 = (const float*)d_in[2];
    const float* w1 = (const float*)d_in[3];
    const float* b1 = (const float*)d_in[4];
    const float* w2 = (const float*)d_in[5];
    const float* b2 = (const float*)d_in[6];
    const float* wr = (const float*)d_in[7];
    const float* br = (const float*)d_in[8];
    float* out = (float*)d_out;

    const int B = in_sizes[0] / in_sizes[1];   // 131072
    const int rows_per_block = ROWS_PER_WAVE * WAVES_PER_BLOCK;  // 128
    const int grid = (B + rows_per_block - 1) / rows_per_block;  // 1024

    ExpressionTree_wmma_kernel<<<grid, BLOCK_THREADS, 0, stream>>>(
        x, c1, c2, w1, b1, w2, b2, wr, br, out);
}